// Loss_62783831933600
// MI455X (gfx1250) — compile-verified
//
#include <hip/hip_runtime.h>
#include <hip/hip_bf16.h>

// ---------------------------------------------------------------------------
// YOLO-style loss, MI455X (gfx1250, wave32).
// Bandwidth-bound: ~367 MB in / 24 B out => ~15.8 us floor @ 23.3 TB/s.
// Strategy: 1 thread per 4 (b,i,j) cells (stride 256), 6x b128 loads for the 24 pred floats,
// all math in registers; per-block reduction of the 5 loss components done
// with V_WMMA_F32_16X16X4_F32 (B = ones => D[m][n] = sum_k A[m][k], exact
// fp32 FMA reduction). Two-pass deterministic reduction via d_ws.
// ---------------------------------------------------------------------------

typedef __attribute__((ext_vector_type(2))) float v2f;
typedef __attribute__((ext_vector_type(8))) float v8f;

__device__ __forceinline__ float fast_sigmoid(float x) {
  return 1.0f / (1.0f + __expf(-x));
}

// Exact fp32 sum of red[comp][0..255] using WMMA with a ones B-matrix.
// Must be called by wave 0 only (threadIdx.x < 32), full EXEC.
// A-matrix 16x4 f32 layout (ISA 7.12.2): VGPR0: lanes0-15 K=0, lanes16-31 K=2;
//                                        VGPR1: lanes0-15 K=1, lanes16-31 K=3.
// With B = ones(4x16): D[m][n] = sum_k A[m][k] (same for every column n), so
// after 4 accumulating WMMAs (4 chunks of 64 values) each lane's 8 D VGPRs
// hold 8 row-sums; lanes 0-15 carry rows 0-7, lanes 16-31 rows 8-15.
__device__ __forceinline__ float block_reduce_wmma(const float (*red)[256],
                                                   int comp, int lane) {
  const int m  = lane & 15;
  const int kb = (lane & 16) ? 2 : 0;
  v8f d = {0.f, 0.f, 0.f, 0.f, 0.f, 0.f, 0.f, 0.f};
  v2f b;
  b.x = 1.0f;
  b.y = 1.0f;
#pragma unroll
  for (int ch = 0; ch < 4; ++ch) {
    v2f a;
    a.x = red[comp][ch * 64 + m * 4 + kb + 0];
    a.y = red[comp][ch * 64 + m * 4 + kb + 1];
    // D = A * ones + D  (exact f32 FMA accumulation)
    d = __builtin_amdgcn_wmma_f32_16x16x4_f32(false, a, false, b, (short)0, d,
                                              false, false);
  }
  float p = 0.0f;
#pragma unroll
  for (int i = 0; i < 8; ++i) p += d[i];
  // lanes 0-15 hold sum(rows 0-7), lanes 16-31 hold sum(rows 8-15); combine.
  p += __shfl_xor(p, 16, 32);
  return p;  // total of all 256 values, replicated across the wave
}

#define CELLS_PER_BLOCK 1024

__global__ __launch_bounds__(256) void yolo_loss_main(
    const float* __restrict__ pred, const float* __restrict__ tgt,
    float* __restrict__ partial, int n_cells, int nblocks) {
  __shared__ float red[5][256];
  const int tid  = threadIdx.x;
  const int base = blockIdx.x * CELLS_PER_BLOCK;

  const float INV_SCALE = 40.0f / 6.5131f;  // 1 / bbox_target_scale

  float a_xy = 0.f, a_wh = 0.f, a_obj = 0.f, a_noobj = 0.f, a_cls = 0.f;

#pragma unroll
  for (int j = 0; j < 4; ++j) {
    const int cell = base + j * 256 + tid;
    if (cell < n_cells) {
      // speculative prefetch of next stripe (global_prefetch_b8; OOB is safe)
      if (j < 3) __builtin_prefetch(pred + (size_t)(cell + 256) * 24, 0, 0);

      const float4 t = ((const float4* __restrict__)tgt)[cell];
      const float4* __restrict__ pp =
          (const float4* __restrict__)(pred + (size_t)cell * 24);
      const float4 q0 = pp[0], q1 = pp[1], q2 = pp[2];
      const float4 q3 = pp[3], q4 = pp[4], q5 = pp[5];

      const bool  coord = (t.z > 0.0f);
      const float cf    = coord ? 1.0f : 0.0f;
      const float nf    = (t.z == 0.0f) ? 1.0f : 0.0f;

      // confidence terms
      const float sig_conf = fast_sigmoid(q0.z);
      const float dc       = sig_conf - t.z;
      a_obj   += cf * dc * dc;
      a_noobj += nf * sig_conf * sig_conf;  // conf_t == 0 on noobj cells

      // xy term
      const float sig_x = fast_sigmoid(q0.x);
      const float dx    = sig_x - t.x;
      a_xy += cf * dx * dx;

      // wh term (w_pred NOT sigmoided; target log(w / SCALE))
      const float wt = coord ? t.y : 1.0f;
      const float dw = q0.y - __logf(wt * INV_SCALE);
      a_wh += cf * dw * dw;

      // class / mass term: softmax(pred[3:24]) . mass, smooth-L1 vs 10
      float cv[21];
      cv[0]  = q0.w;
      cv[1]  = q1.x; cv[2]  = q1.y; cv[3]  = q1.z; cv[4]  = q1.w;
      cv[5]  = q2.x; cv[6]  = q2.y; cv[7]  = q2.z; cv[8]  = q2.w;
      cv[9]  = q3.x; cv[10] = q3.y; cv[11] = q3.z; cv[12] = q3.w;
      cv[13] = q4.x; cv[14] = q4.y; cv[15] = q4.z; cv[16] = q4.w;
      cv[17] = q5.x; cv[18] = q5.y; cv[19] = q5.z; cv[20] = q5.w;

      float mx = cv[0];
#pragma unroll
      for (int c = 1; c < 21; ++c) mx = fmaxf(mx, cv[c]);
      float se = 0.f, sm = 0.f;
#pragma unroll
      for (int c = 0; c < 21; ++c) {
        const float e = __expf(cv[c] - mx);
        se += e;
        sm += e * (1.0f + 0.5f * (float)c);  // mass bins 1.0 .. 11.0
      }
      const float pm = sm / se;

      const bool  cmask  = coord && (t.w > 0.0f);
      const float ctsafe = cmask ? t.w : 1.0f;
      const float diff   = 10.0f * (pm + 1.0f) / (ctsafe + 1.0f) - 10.0f;
      const float ad     = fabsf(diff);
      const float sl1    = (ad < 1.0f) ? 0.5f * diff * diff : (ad - 0.5f);
      a_cls += cmask ? sl1 : 0.0f;
    }
  }

  red[0][tid] = a_xy;
  red[1][tid] = a_wh;
  red[2][tid] = a_obj;
  red[3][tid] = a_noobj;
  red[4][tid] = a_cls;
  __syncthreads();

  if (tid < 32) {  // wave 0, full EXEC within the wave
#pragma unroll
    for (int c = 0; c < 5; ++c) {
      const float tot = block_reduce_wmma(red, c, tid);
      if (tid == 0) partial[c * nblocks + blockIdx.x] = tot;
    }
  }
}

__global__ __launch_bounds__(256) void yolo_loss_final(
    const float* __restrict__ partial, float* __restrict__ out, int nblocks,
    float inv_bs) {
  __shared__ float red[5][256];
  __shared__ float tot[5];
  const int tid = threadIdx.x;

#pragma unroll
  for (int c = 0; c < 5; ++c) {
    float s = 0.f;
    for (int i = tid; i < nblocks; i += 256) s += partial[c * nblocks + i];
    red[c][tid] = s;
  }
  __syncthreads();

  if (tid < 32) {
#pragma unroll
    for (int c = 0; c < 5; ++c) {
      const float t = block_reduce_wmma(red, c, tid);
      if (tid == 0) tot[c] = t;
    }
  }
  __syncthreads();

  if (tid == 0) {
    const float xy = tot[0], wh = tot[1], obj = tot[2];
    const float noobj = tot[3], cls = tot[4];
    const float loss = 10.0f * (xy + wh) + obj + 1.0f * noobj + 0.5f * cls;
    out[0] = xy * inv_bs;
    out[1] = wh * inv_bs;
    out[2] = obj * inv_bs;
    out[3] = noobj * inv_bs;
    out[4] = cls * inv_bs;
    out[5] = loss * inv_bs;
  }
}

extern "C" void kernel_launch(void* const* d_in, const int* in_sizes, int n_in,
                              void* d_out, int out_size, void* d_ws,
                              size_t ws_size, hipStream_t stream) {
  const float* pred = (const float*)d_in[0];
  const float* tgt  = (const float*)d_in[1];
  float* out        = (float*)d_out;

  const int n_cells = in_sizes[1] / 4;              // BATCH * S * S
  const int bs      = in_sizes[0] / (10 * 10 * 24); // BATCH
  const int nblocks = (n_cells + CELLS_PER_BLOCK - 1) / CELLS_PER_BLOCK;

  float* partial = (float*)d_ws;  // [5][nblocks] floats, fully written pass 1

  yolo_loss_main<<<nblocks, 256, 0, stream>>>(pred, tgt, partial, n_cells,
                                              nblocks);
  yolo_loss_final<<<1, 256, 0, stream>>>(partial, out, nblocks,
                                         1.0f / (float)bs);
}